// AVIN_6794638262657
// MI455X (gfx1250) — compile-verified
//
#include <hip/hip_runtime.h>
#include <hip/hip_bf16.h>
#include <math.h>

typedef __attribute__((ext_vector_type(2))) float v2f;
typedef __attribute__((ext_vector_type(8))) float v8f;

#define Bsz 192
#define Cch 512
#define HW  784
#define KTOP 23
#define INV_TS (1.0f/0.03f)
#define INV_TC (1.0f/0.07f)

// ---------- helpers ----------
__device__ __forceinline__ unsigned f2ord(float f) {
    unsigned u = __float_as_uint(f);
    return (u & 0x80000000u) ? ~u : (u | 0x80000000u);
}
__device__ __forceinline__ float ord2f(unsigned u) {
    unsigned b = (u & 0x80000000u) ? (u & 0x7FFFFFFFu) : ~u;
    return __uint_as_float(b);
}
__device__ __forceinline__ float wredf(float v) {
    #pragma unroll
    for (int o = 16; o; o >>= 1) v += __shfl_xor(v, o);
    return v;
}
__device__ __forceinline__ int wredi(int v) {
    #pragma unroll
    for (int o = 16; o; o >>= 1) v += __shfl_xor(v, o);
    return v;
}

// ---------- K0: audio MLP  fa = relu(ea@Wa1.T)@Wa2.T ----------
__global__ void __launch_bounds__(256) k0_audio(const float* __restrict__ ea,
                                                const float* __restrict__ Wa1,
                                                const float* __restrict__ Wa2,
                                                float* __restrict__ FA) {
    int b = blockIdx.x, tid = threadIdx.x;
    __shared__ float eas[2048];
    __shared__ float h[512];
    for (int i = tid; i < 2048; i += 256) eas[i] = ea[b * 2048 + i];
    __syncthreads();
    for (int j = tid; j < 512; j += 256) {
        const float* wr = Wa1 + (size_t)j * 2048;
        float s = 0.f;
        for (int k = 0; k < 2048; ++k) s += eas[k] * wr[k];
        h[j] = fmaxf(s, 0.f);
    }
    __syncthreads();
    for (int o = tid; o < 512; o += 256) {
        const float* wr = Wa2 + (size_t)o * 512;
        float s = 0.f;
        for (int j = 0; j < 512; ++j) s += h[j] * wr[j];
        FA[b * 512 + o] = s;
    }
}

// ---------- K1: fv_map[b,o,p] = sum_c Wv[o,c] * ev[b,c,p]  (WMMA f32 16x16x4) ----------
// grid (49 p-tiles, 192 b), 256 threads = 8 waves; each wave owns 4 o-tiles (covers all 512 o)
// ev K-panels staged into double-buffered LDS with async global->LDS (ASYNCcnt) so the
// next chunk's HBM fetch overlaps the current chunk's WMMA work.
__global__ void __launch_bounds__(256) k1_gemm1(const float* __restrict__ ev,
                                                const float* __restrict__ Wv,
                                                float* __restrict__ FV) {
    int p0 = blockIdx.x << 4;
    int b  = blockIdx.y;
    int tid = threadIdx.x, lane = tid & 31, w = tid >> 5;
    __shared__ __align__(16) float evs[2][1024];  // 64 channels x 16 pixels, x2 buffers

    int m    = lane & 15;
    int ksel = (lane >> 4) << 1;     // 0 or 2
    int mhi  = (lane >> 4) << 3;     // 0 or 8
    v8f acc[4] = {};

    const float* evb = ev + (size_t)b * Cch * HW;

    // per-thread staging: one b128 (4 floats) per chunk; evs[i], i=4*tid..4*tid+3
    // maps to channel kl = tid>>2, pixels pp..pp+3 with pp = (tid&3)*4
    int kl = tid >> 2;
    int pp = (tid & 3) << 2;
    unsigned ldsA = (unsigned)(size_t)&evs[0][tid << 2];
    unsigned ldsB = (unsigned)(size_t)&evs[1][tid << 2];
    const float* g0 = evb + (size_t)kl * HW + p0 + pp;

    // prologue: chunk 0 -> buffer 0
    asm volatile("global_load_async_to_lds_b128 %0, %1, off"
                 :: "v"(ldsA), "v"(g0) : "memory");

    for (int c = 0; c < 8; ++c) {
        int kk = c << 6;
        if (c + 1 < 8) {
            const float* gn = g0 + (size_t)(kk + 64) * HW;
            unsigned dst = ((c + 1) & 1) ? ldsB : ldsA;
            asm volatile("global_load_async_to_lds_b128 %0, %1, off"
                         :: "v"(dst), "v"(gn) : "memory");
            asm volatile("s_wait_asynccnt 0x1" ::: "memory"); // chunk c done, c+1 in flight
        } else {
            asm volatile("s_wait_asynccnt 0x0" ::: "memory");
        }
        __syncthreads();                 // chunk c visible to all waves
        const float* cb = evs[c & 1];
        #pragma unroll 4
        for (int kb = 0; kb < 64; kb += 4) {
            v2f bv;
            bv.x = cb[(kb + ksel) * 16 + m];
            bv.y = cb[(kb + ksel + 1) * 16 + m];
            #pragma unroll
            for (int t = 0; t < 4; ++t) {
                int o0 = (w + (t << 3)) << 4;
                const float* ap = Wv + (size_t)(o0 + m) * Cch + kk + kb + ksel;
                v2f av; av.x = ap[0]; av.y = ap[1];
                acc[t] = __builtin_amdgcn_wmma_f32_16x16x4_f32(
                    false, av, false, bv, (short)0, acc[t], false, false);
            }
        }
        __syncthreads();                 // buffer c&1 free before chunk c+2 lands
    }
    #pragma unroll
    for (int t = 0; t < 4; ++t) {
        int ob = (((w + (t << 3)) << 4)) + mhi;
        #pragma unroll
        for (int r = 0; r < 8; ++r)
            FV[((size_t)(b * Cch + ob + r)) * HW + p0 + m] = acc[t][r];
    }
}

// ---------- K2: per-pixel inv-norm over channels + ind_vec channel means ----------
__global__ void __launch_bounds__(256) k2_reduce(const float* __restrict__ FV,
                                                 float* __restrict__ INVN,
                                                 float* __restrict__ INDV) {
    int b = blockIdx.x, tid = threadIdx.x, lane = tid & 31, w = tid >> 5;
    __shared__ float osum[512 * 8];
    float ssq[4] = {0.f, 0.f, 0.f, 0.f};
    int np = (tid < 16) ? 4 : 3;           // 784 = 3*256 + 16
    for (int o = 0; o < Cch; ++o) {
        const float* row = FV + ((size_t)(b * Cch + o)) * HW;
        float cs = 0.f;
        for (int t = 0; t < np; ++t) {
            int p = tid + t * 256;
            float v = row[p];
            ssq[t] += v * v; cs += v;
        }
        cs = wredf(cs);
        if (lane == 0) osum[o * 8 + w] = cs;
    }
    __syncthreads();
    for (int o = tid; o < Cch; o += 256) {
        float s = 0.f;
        #pragma unroll
        for (int q = 0; q < 8; ++q) s += osum[o * 8 + q];
        INDV[b * Cch + o] = s * (1.f / (float)HW);
    }
    for (int t = 0; t < np; ++t) {
        int p = tid + t * 256;
        INVN[b * HW + p] = 1.f / fmaxf(sqrtf(ssq[t]), 1e-12f);
    }
}

// ---------- K3: iv_norm (row-major + transposed) ----------
__global__ void __launch_bounds__(256) k3_norm(const float* __restrict__ INDV,
                                               float* __restrict__ IVN,
                                               float* __restrict__ IVNT) {
    int d = blockIdx.x, tid = threadIdx.x;
    __shared__ float red[256];
    float v0 = INDV[d * Cch + tid];
    float v1 = INDV[d * Cch + tid + 256];
    red[tid] = v0 * v0 + v1 * v1;
    __syncthreads();
    for (int s = 128; s; s >>= 1) { if (tid < s) red[tid] += red[tid + s]; __syncthreads(); }
    float inv = 1.f / fmaxf(sqrtf(red[0]), 1e-12f);
    IVN[d * Cch + tid]        = v0 * inv;
    IVN[d * Cch + tid + 256]  = v1 * inv;
    IVNT[tid * Bsz + d]         = v0 * inv;
    IVNT[(tid + 256) * Bsz + d] = v1 * inv;
}

// ---------- K4: fused S-tile GEMM (WMMA) + exact top-k thresholds + soft-mask means ----------
// grid (12 d-chunks of 16, 192 b), 256 threads = 8 waves
__global__ void __launch_bounds__(256) k4_sim(const float* __restrict__ FV,
                                              const float* __restrict__ INVN,
                                              const float* __restrict__ IVNT,
                                              float* __restrict__ SP,
                                              float* __restrict__ SN) {
    int d0 = blockIdx.x << 4;
    int b  = blockIdx.y;
    int tid = threadIdx.x, lane = tid & 31, w = tid >> 5;
    __shared__ float Bsm[Cch * 16];      // iv_norm^T panel [k][dl]
    __shared__ float Sl[HW * 16];        // S tile [p][dl]

    for (int i = tid; i < Cch * 16; i += 256) {
        int k = i >> 4, dl = i & 15;
        Bsm[i] = IVNT[k * Bsz + d0 + dl];
    }
    __syncthreads();

    int m    = lane & 15;
    int ksel = (lane >> 4) << 1;
    int mhi  = (lane >> 4) << 3;
    const float* fvb  = FV + (size_t)b * Cch * HW;
    const float* invb = INVN + b * HW;

    for (int pt = w; pt < 49; pt += 8) {
        int p0 = pt << 4;
        float rn = invb[p0 + m];         // per-row (pixel) inverse norm
        v8f acc = {};
        for (int k = 0; k < Cch; k += 4) {
            const float* ap = fvb + (size_t)(k + ksel) * HW + p0 + m;
            v2f av; av.x = ap[0] * rn; av.y = ap[HW] * rn;
            v2f bv;
            bv.x = Bsm[(k + ksel) * 16 + m];
            bv.y = Bsm[(k + ksel + 1) * 16 + m];
            acc = __builtin_amdgcn_wmma_f32_16x16x4_f32(
                false, av, false, bv, (short)0, acc, false, false);
        }
        #pragma unroll
        for (int r = 0; r < 8; ++r)
            Sl[(p0 + r + mhi) * 16 + m] = acc[r];
    }
    __syncthreads();

    // Phase 2: each wave owns 2 of the 16 d columns
    for (int dd = 0; dd < 2; ++dd) {
        int dl = w * 2 + dd;
        bool has25 = (lane < 16);        // 784 = 24*32 + 16
        float vals[25];
        unsigned uv[25];
        #pragma unroll
        for (int t = 0; t < 25; ++t) {
            bool ok = (t < 24) || has25;
            float x = ok ? Sl[(lane + t * 32) * 16 + dl] : 0.f;
            vals[t] = x;
            uv[t] = f2ord(x);
        }
        // 23rd largest: max t with |{u >= t}| >= 23 (monotone binary search on ordered uints)
        unsigned lo = 0u, hi = 0xFFFFFFFFu;
        while (lo < hi) {
            unsigned mid = lo + ((hi - lo) >> 1) + 1u;
            int c = 0;
            #pragma unroll
            for (int t = 0; t < 24; ++t) c += (uv[t] >= mid) ? 1 : 0;
            if (has25) c += (uv[24] >= mid) ? 1 : 0;
            c = wredi(c);
            if (c >= KTOP) lo = mid; else hi = mid - 1u;
        }
        float posthr = ord2f(lo);
        // 23rd smallest: min t with |{u <= t}| >= 23
        lo = 0u; hi = 0xFFFFFFFFu;
        while (lo < hi) {
            unsigned mid = lo + ((hi - lo) >> 1);
            int c = 0;
            #pragma unroll
            for (int t = 0; t < 24; ++t) c += (uv[t] <= mid) ? 1 : 0;
            if (has25) c += (uv[24] <= mid) ? 1 : 0;
            c = wredi(c);
            if (c >= KTOP) hi = mid; else lo = mid + 1u;
        }
        float negthr = ord2f(lo);

        float spn = 0.f, spd = 0.f, snn = 0.f, snd = 0.f;
        #pragma unroll
        for (int t = 0; t < 24; ++t) {
            float x = vals[t];
            float mp = 1.f / (1.f + __expf(-(x - posthr) * INV_TS));
            float mn = 1.f - 1.f / (1.f + __expf(-(x - negthr) * INV_TS));
            spn += x * mp; spd += mp; snn += x * mn; snd += mn;
        }
        if (has25) {
            float x = vals[24];
            float mp = 1.f / (1.f + __expf(-(x - posthr) * INV_TS));
            float mn = 1.f - 1.f / (1.f + __expf(-(x - negthr) * INV_TS));
            spn += x * mp; spd += mp; snn += x * mn; snd += mn;
        }
        spn = wredf(spn); spd = wredf(spd); snn = wredf(snn); snd = wredf(snd);
        if (lane == 0) {
            SP[b * Bsz + d0 + dl] = spn / spd;
            SN[b * Bsz + d0 + dl] = snn / snd;
        }
    }
}

// ---------- K5: contrastive CE losses (rows + cols) ----------
__global__ void __launch_bounds__(256) k5_ce(const float* __restrict__ SP,
                                             const float* __restrict__ SN,
                                             float* __restrict__ ACC) {
    int b = blockIdx.x, tid = threadIdx.x;
    __shared__ float red[256];
    const float NEG = -1e30f;

    // row1 = [SP[b,:], SN[b,:]]/TC
    float l1a = ((tid < 192) ? SP[b * Bsz + tid] : SN[b * Bsz + (tid - 192)]) * INV_TC;
    float l1b = (tid < 128) ? SN[b * Bsz + (tid + 64)] * INV_TC : NEG;
    red[tid] = fmaxf(l1a, l1b); __syncthreads();
    for (int s = 128; s; s >>= 1) { if (tid < s) red[tid] = fmaxf(red[tid], red[tid + s]); __syncthreads(); }
    float M1 = red[0]; __syncthreads();
    red[tid] = __expf(l1a - M1) + ((tid < 128) ? __expf(l1b - M1) : 0.f); __syncthreads();
    for (int s = 128; s; s >>= 1) { if (tid < s) red[tid] += red[tid + s]; __syncthreads(); }
    float lse1 = M1 + __logf(red[0]); __syncthreads();

    // row2 = [SP[:,b], SN[:,b]]/TC
    float l2a = ((tid < 192) ? SP[tid * Bsz + b] : SN[(tid - 192) * Bsz + b]) * INV_TC;
    float l2b = (tid < 128) ? SN[(tid + 64) * Bsz + b] * INV_TC : NEG;
    red[tid] = fmaxf(l2a, l2b); __syncthreads();
    for (int s = 128; s; s >>= 1) { if (tid < s) red[tid] = fmaxf(red[tid], red[tid + s]); __syncthreads(); }
    float M2 = red[0]; __syncthreads();
    red[tid] = __expf(l2a - M2) + ((tid < 128) ? __expf(l2b - M2) : 0.f); __syncthreads();
    for (int s = 128; s; s >>= 1) { if (tid < s) red[tid] += red[tid + s]; __syncthreads(); }
    float lse2 = M2 + __logf(red[0]);

    if (tid == 0) {
        float diag = SP[b * Bsz + b] * INV_TC;
        atomicAdd(&ACC[0], lse1 - diag);
        atomicAdd(&ACC[1], lse2 - diag);
    }
}

// ---------- K6: pairwise-distance loss row/col sums ----------
__global__ void __launch_bounds__(256) k6_dist(const float* __restrict__ INDV,
                                               const float* __restrict__ FA,
                                               const float* __restrict__ IVN,
                                               float* __restrict__ ROW,
                                               float* __restrict__ COL) {
    int gw = (int)((blockIdx.x * blockDim.x + threadIdx.x) >> 5);
    int lane = threadIdx.x & 31;
    if (gw >= Bsz * Bsz) return;
    int i = gw / Bsz, j = gw % Bsz;
    float ds = 0.f, vv = 0.f;
    for (int c = lane; c < Cch; c += 32) {
        float dd = INDV[i * Cch + c] - FA[j * Cch + c] + 1e-6f;
        ds += dd * dd;
        vv += IVN[i * Cch + c] * IVN[j * Cch + c];
    }
    ds = wredf(ds); vv = wredf(vv);
    if (lane == 0) {
        float wgt = (i == j) ? 1.f : -vv;
        float msk = (i == j) ? 1.f : (-1.f / (float)(Bsz - 1));
        float val = ds * wgt * msk;
        atomicAdd(&ROW[i], val);
        atomicAdd(&COL[j], val);
    }
}

// ---------- K7: final losses ----------
__global__ void __launch_bounds__(256) k7_final(const float* __restrict__ ACC,
                                                const float* __restrict__ ROW,
                                                const float* __restrict__ COL,
                                                float* __restrict__ out) {
    int tid = threadIdx.x;
    __shared__ float r1[256], r2[256];
    r1[tid] = (tid < Bsz) ? fmaxf(ROW[tid] + 0.6f, 0.f) : 0.f;
    r2[tid] = (tid < Bsz) ? fmaxf(COL[tid] + 0.6f, 0.f) : 0.f;
    __syncthreads();
    for (int s = 128; s; s >>= 1) {
        if (tid < s) { r1[tid] += r1[tid + s]; r2[tid] += r2[tid + s]; }
        __syncthreads();
    }
    if (tid == 0) {
        out[0] = (ACC[0] + ACC[1]) * (0.5f / (float)Bsz);
        out[1] = (r1[0] + r2[0]) * (0.5f / (float)Bsz);
    }
}

// ---------- workspace layout (floats) ----------
static const size_t OFF_FV   = 0;                                   // 192*512*784
static const size_t OFF_FA   = OFF_FV + (size_t)Bsz * Cch * HW;     // 192*512
static const size_t OFF_INVN = OFF_FA + (size_t)Bsz * Cch;          // 192*784
static const size_t OFF_INDV = OFF_INVN + (size_t)Bsz * HW;         // 192*512
static const size_t OFF_IVN  = OFF_INDV + (size_t)Bsz * Cch;        // 192*512
static const size_t OFF_IVNT = OFF_IVN + (size_t)Bsz * Cch;         // 512*192
static const size_t OFF_SP   = OFF_IVNT + (size_t)Cch * Bsz;        // 192*192
static const size_t OFF_SN   = OFF_SP + (size_t)Bsz * Bsz;          // 192*192
static const size_t OFF_ACC  = OFF_SN + (size_t)Bsz * Bsz;          // 2
static const size_t OFF_ROW  = OFF_ACC + 2;                         // 192
static const size_t OFF_COL  = OFF_ROW + Bsz;                       // 192

extern "C" void kernel_launch(void* const* d_in, const int* in_sizes, int n_in,
                              void* d_out, int out_size, void* d_ws, size_t ws_size,
                              hipStream_t stream) {
    const float* ev  = (const float*)d_in[0];
    const float* ea  = (const float*)d_in[1];
    const float* Wv  = (const float*)d_in[2];
    const float* Wa1 = (const float*)d_in[3];
    const float* Wa2 = (const float*)d_in[4];
    float* ws = (float*)d_ws;

    float* FV   = ws + OFF_FV;
    float* FA   = ws + OFF_FA;
    float* INVN = ws + OFF_INVN;
    float* INDV = ws + OFF_INDV;
    float* IVN  = ws + OFF_IVN;
    float* IVNT = ws + OFF_IVNT;
    float* SP   = ws + OFF_SP;
    float* SN   = ws + OFF_SN;
    float* ACC  = ws + OFF_ACC;
    float* ROW  = ws + OFF_ROW;
    float* COL  = ws + OFF_COL;

    hipMemsetAsync(ACC, 0, (2 + Bsz + Bsz) * sizeof(float), stream);

    k0_audio<<<Bsz, 256, 0, stream>>>(ea, Wa1, Wa2, FA);
    k1_gemm1<<<dim3(HW / 16, Bsz), 256, 0, stream>>>(ev, Wv, FV);
    k2_reduce<<<Bsz, 256, 0, stream>>>(FV, INVN, INDV);
    k3_norm<<<Bsz, 256, 0, stream>>>(INDV, IVN, IVNT);
    k4_sim<<<dim3(Bsz / 16, Bsz), 256, 0, stream>>>(FV, INVN, IVNT, SP, SN);
    k5_ce<<<Bsz, 256, 0, stream>>>(SP, SN, ACC);
    k6_dist<<<(Bsz * Bsz) / 8, 256, 0, stream>>>(INDV, FA, IVN, ROW, COL);
    k7_final<<<1, 256, 0, stream>>>(ACC, ROW, COL, (float*)d_out);
}